// CurvatureEnhancedGAT_19559281066282
// MI455X (gfx1250) — compile-verified
//
#include <hip/hip_runtime.h>

typedef __attribute__((ext_vector_type(2))) float v2f;
typedef __attribute__((ext_vector_type(8))) float v8f;

#define FEAT 128
#define NEG_SLOPE 0.2f
#define EPS_ 1e-16f
#define GAMMA_ 0.3f

// ---------------------------------------------------------------------------
// order-independent float atomic max (sign-split trick); init value is -inf
// ---------------------------------------------------------------------------
__device__ __forceinline__ void atomicMaxF(float* addr, float v) {
    if (v >= 0.0f) atomicMax((int*)addr, __float_as_int(v));
    else           atomicMin((unsigned int*)addr, __float_as_uint(v));
}

// ---------------------------------------------------------------------------
// scale = 1 + GAMMA * mean(curvature)  — deterministic 2-stage reduction
// ---------------------------------------------------------------------------
__global__ void reduce_partial(const float* __restrict__ w, float* partials, int n) {
    __shared__ float sm[256];
    int t = threadIdx.x;
    float s = 0.0f;
    for (int i = blockIdx.x * 256 + t; i < n; i += gridDim.x * 256) s += w[i];
    sm[t] = s; __syncthreads();
    for (int off = 128; off > 0; off >>= 1) {
        if (t < off) sm[t] += sm[t + off];
        __syncthreads();
    }
    if (t == 0) partials[blockIdx.x] = sm[0];
}

__global__ void reduce_final(const float* __restrict__ partials, float* scale_p,
                             int nb, int n) {
    __shared__ float sm[256];
    int t = threadIdx.x;
    float s = 0.0f;
    for (int i = t; i < nb; i += 256) s += partials[i];
    sm[t] = s; __syncthreads();
    for (int off = 128; off > 0; off >>= 1) {
        if (t < off) sm[t] += sm[t + off];
        __syncthreads();
    }
    if (t == 0) scale_p[0] = 1.0f + GAMMA_ * (sm[0] / (float)n);
}

// ---------------------------------------------------------------------------
// H[nrows,128] = X[nrows,128] @ W[128,128]   via V_WMMA_F32_16X16X4_F32
// block = 256 threads = 8 waves; wave w owns rows [blk*128 + w*16, +16)
// W staged into LDS once per block (64 KB of the 320 KB/WGP)
// ---------------------------------------------------------------------------
__global__ __launch_bounds__(256) void gemm128_wmma(const float* __restrict__ X,
                                                    const float* __restrict__ W,
                                                    float* __restrict__ H, int nrows) {
    __shared__ float sW[FEAT * FEAT];
    const int t = threadIdx.x;
    {
        const float4* src = (const float4*)W;
        float4* dst = (float4*)sW;
        for (int i = t; i < FEAT * FEAT / 4; i += 256) dst[i] = src[i];
    }
    __syncthreads();

    const int wave  = t >> 5;
    const int lane  = t & 31;
    const int lhalf = lane >> 4;   // 0: lanes 0-15, 1: lanes 16-31
    const int l15   = lane & 15;
    const int m0    = blockIdx.x * 128 + wave * 16;

    int arow = m0 + l15;
    if (arow >= nrows) arow = nrows - 1;          // clamp; stores are predicated
    const float* Xrow = X + (size_t)arow * FEAT;

    for (int nt = 0; nt < 8; ++nt) {
        const int n0 = nt * 16;
        v8f acc = {};
#pragma unroll
        for (int kk = 0; kk < 32; ++kk) {
            const int k0 = kk * 4 + lhalf * 2;    // ISA 16x4 A layout: K pairs per half
            v2f a, b;
            a.x = Xrow[k0];
            a.y = Xrow[k0 + 1];
            b.x = sW[(k0)     * FEAT + n0 + l15]; // B 4x16: row K striped across lanes
            b.y = sW[(k0 + 1) * FEAT + n0 + l15];
            acc = __builtin_amdgcn_wmma_f32_16x16x4_f32(
                false, a, false, b, (short)0, acc, false, false);
        }
#pragma unroll
        for (int g = 0; g < 8; ++g) {             // C/D: VGPR g -> M = g + 8*lhalf
            const int row = m0 + lhalf * 8 + g;
            if (row < nrows) H[(size_t)row * FEAT + n0 + l15] = acc[g];
        }
    }
}

// ---------------------------------------------------------------------------
// per-(node,head): attention scores + init m=-inf, denom=0, zero agg slice
// ---------------------------------------------------------------------------
__global__ void node_prep(const float* __restrict__ H,
                          const float* __restrict__ a_src,
                          const float* __restrict__ a_dst,
                          float* __restrict__ asn, float* __restrict__ adn,
                          float* __restrict__ mb, float* __restrict__ db,
                          float* __restrict__ agg,
                          int N_, int heads, int ch) {
    int idx = blockIdx.x * blockDim.x + threadIdx.x;   // n*heads + h
    if (idx >= N_ * heads) return;
    int n = idx / heads, h = idx - n * heads;
    const float* hp  = H + (size_t)n * FEAT + h * ch;
    const float* avs = a_src + h * ch;
    const float* avd = a_dst + h * ch;
    float s0 = 0.0f, s1 = 0.0f;
    for (int c = 0; c < ch; ++c) {
        float v = hp[c];
        s0 += v * avs[c];
        s1 += v * avd[c];
    }
    asn[idx] = s0;
    adn[idx] = s1;
    mb[idx]  = -__builtin_inff();
    db[idx]  = 0.0f;
    float* ag = agg + (size_t)n * FEAT + h * ch;
    for (int c = 0; c < ch; ++c) ag[c] = 0.0f;
}

__device__ __forceinline__ void edge_ends(int e, int E_, const int* __restrict__ esrc,
                                          const int* __restrict__ edst, int& s, int& d) {
    if (e < E_) { s = esrc[e]; d = edst[e]; }
    else        { s = d = e - E_; }                // appended self-loops
}

// ---------------------------------------------------------------------------
// pass 1: segment max of leaky_relu(as[src]+ad[dst]) per (dst, head)
// ---------------------------------------------------------------------------
__global__ void edge_max(const int* __restrict__ esrc, const int* __restrict__ edst,
                         const float* __restrict__ asn, const float* __restrict__ adn,
                         float* __restrict__ mb, int E_, int Etot, int heads) {
    int idx = blockIdx.x * blockDim.x + threadIdx.x;   // e*heads + h
    if (idx >= Etot * heads) return;
    int e = idx / heads, h = idx - e * heads;
    int s, d; edge_ends(e, E_, esrc, edst, s, d);
    float ev = asn[s * heads + h] + adn[d * heads + h];
    ev = ev > 0.0f ? ev : NEG_SLOPE * ev;
    atomicMaxF(&mb[d * heads + h], ev);
}

// ---------------------------------------------------------------------------
// pass 2: denom[dst,h] += exp(e - m[dst,h])   (m clamped to 0 if non-finite)
// ---------------------------------------------------------------------------
__global__ void edge_sum(const int* __restrict__ esrc, const int* __restrict__ edst,
                         const float* __restrict__ asn, const float* __restrict__ adn,
                         const float* __restrict__ mb, float* __restrict__ db,
                         int E_, int Etot, int heads) {
    int idx = blockIdx.x * blockDim.x + threadIdx.x;
    if (idx >= Etot * heads) return;
    int e = idx / heads, h = idx - e * heads;
    int s, d; edge_ends(e, E_, esrc, edst, s, d);
    float ev = asn[s * heads + h] + adn[d * heads + h];
    ev = ev > 0.0f ? ev : NEG_SLOPE * ev;
    float mc = mb[d * heads + h];
    if (!(mc >= -3.0e38f && mc <= 3.0e38f)) mc = 0.0f;
    atomicAdd(&db[d * heads + h], __expf(ev - mc));
}

// ---------------------------------------------------------------------------
// pass 3: agg[dst,f] += h[src,f] * alpha   — one wave32 per edge,
// lane l covers features l, l+32, l+64, l+96 (coalesced 128B per wave per k)
// ---------------------------------------------------------------------------
__global__ void edge_aggr(const int* __restrict__ esrc, const int* __restrict__ edst,
                          const float* __restrict__ H,
                          const float* __restrict__ asn, const float* __restrict__ adn,
                          const float* __restrict__ mb, const float* __restrict__ db,
                          float* __restrict__ agg,
                          int E_, int Etot, int heads, int ch) {
    int gid  = blockIdx.x * blockDim.x + threadIdx.x;
    int e    = gid >> 5;
    int lane = gid & 31;
    if (e >= Etot) return;
    int s, d; edge_ends(e, E_, esrc, edst, s, d);

    float alpha[4];
    for (int h = 0; h < heads; ++h) {
        float ev = asn[s * heads + h] + adn[d * heads + h];
        ev = ev > 0.0f ? ev : NEG_SLOPE * ev;
        float mc = mb[d * heads + h];
        if (!(mc >= -3.0e38f && mc <= 3.0e38f)) mc = 0.0f;
        alpha[h] = __expf(ev - mc) / (db[d * heads + h] + EPS_);
    }

    const float* hs = H   + (size_t)s * FEAT;
    float*       ag = agg + (size_t)d * FEAT;
#pragma unroll
    for (int k = 0; k < 4; ++k) {
        int f = k * 32 + lane;
        int h = f / ch;                       // ch=32 -> head k ; ch=128 -> head 0
        atomicAdd(&ag[f], hs[f] * alpha[h]);
    }
}

// ---------------------------------------------------------------------------
// out = relu(scale*(agg + b))  (layers 1,2)   or   agg + b  (layer 3)
// ---------------------------------------------------------------------------
__global__ void post_act(float* __restrict__ io, const float* __restrict__ bias,
                         const float* __restrict__ scale_p, int total, int relu) {
    int i = blockIdx.x * blockDim.x + threadIdx.x;
    if (i >= total) return;
    float v = io[i] + bias[i & (FEAT - 1)];
    if (relu) {
        v *= scale_p[0];
        v = v > 0.0f ? v : 0.0f;
    }
    io[i] = v;
}

// ---------------------------------------------------------------------------
extern "C" void kernel_launch(void* const* d_in, const int* in_sizes, int n_in,
                              void* d_out, int out_size, void* d_ws, size_t ws_size,
                              hipStream_t stream) {
    const float* x     = (const float*)d_in[0];
    const int*   ei    = (const int*)  d_in[1];
    const float* curv  = (const float*)d_in[2];
    const float* W1    = (const float*)d_in[3];
    const float* as1   = (const float*)d_in[4];
    const float* ad1   = (const float*)d_in[5];
    const float* b1    = (const float*)d_in[6];
    const float* W2    = (const float*)d_in[7];
    const float* as2   = (const float*)d_in[8];
    const float* ad2   = (const float*)d_in[9];
    const float* b2    = (const float*)d_in[10];
    const float* W3    = (const float*)d_in[11];
    const float* as3   = (const float*)d_in[12];
    const float* ad3   = (const float*)d_in[13];
    const float* b3    = (const float*)d_in[14];

    const int N_   = in_sizes[0] / FEAT;
    const int E_   = in_sizes[1] / 2;
    const int Etot = E_ + N_;
    const int* esrc = ei;
    const int* edst = ei + E_;

    float* ws       = (float*)d_ws;
    float* scale_p  = ws;                                  // [1]
    float* partials = ws + 16;                             // [1024]
    float* bufA     = ws + 2048;                           // [N,128] GEMM out
    float* bufB     = bufA + (size_t)N_ * FEAT;            // [N,128] aggregate
    float* asn      = bufB + (size_t)N_ * FEAT;            // [N,4]
    float* adn      = asn  + (size_t)N_ * 4;               // [N,4]
    float* mb       = adn  + (size_t)N_ * 4;               // [N,4]
    float* db       = mb   + (size_t)N_ * 4;               // [N,4]

    // curvature scale
    reduce_partial<<<1024, 256, 0, stream>>>(curv, partials, E_);
    reduce_final  <<<1,    256, 0, stream>>>(partials, scale_p, 1024, E_);

    const unsigned gGemm = (unsigned)((N_ + 127) / 128);
    const unsigned gPost = (unsigned)(((size_t)N_ * FEAT + 255) / 256);
    const unsigned gAggr = (unsigned)(((long long)Etot * 32 + 255) / 256);

    auto layer = [&](const float* Xin, const float* Wm, const float* a_s,
                     const float* a_d, const float* bias, float* agg,
                     int heads, int ch, int relu) {
        const unsigned gNode = (unsigned)((N_ * heads + 255) / 256);
        const unsigned gEdge = (unsigned)(((long long)Etot * heads + 255) / 256);
        gemm128_wmma<<<gGemm, 256, 0, stream>>>(Xin, Wm, bufA, N_);
        node_prep   <<<gNode, 256, 0, stream>>>(bufA, a_s, a_d, asn, adn, mb, db,
                                                agg, N_, heads, ch);
        edge_max    <<<gEdge, 256, 0, stream>>>(esrc, edst, asn, adn, mb,
                                                E_, Etot, heads);
        edge_sum    <<<gEdge, 256, 0, stream>>>(esrc, edst, asn, adn, mb, db,
                                                E_, Etot, heads);
        edge_aggr   <<<gAggr, 256, 0, stream>>>(esrc, edst, bufA, asn, adn, mb, db,
                                                agg, E_, Etot, heads, ch);
        post_act    <<<gPost, 256, 0, stream>>>(agg, bias, scale_p, N_ * FEAT, relu);
    };

    layer(x,    W1, as1, ad1, b1, bufB,           4,  32, 1);
    layer(bufB, W2, as2, ad2, b2, bufB,           4,  32, 1);   // gemm drains bufB first
    layer(bufB, W3, as3, ad3, b3, (float*)d_out,  1, 128, 0);
}